// _multiCodebookQuantization_44573170597902
// MI455X (gfx1250) — compile-verified
//
#include <hip/hip_runtime.h>
#include <math.h>

// fp32 WMMA fragment types (wave32): A/B 16x4 f32 = 2 VGPRs, C/D 16x16 f32 = 8 VGPRs
typedef float v2f __attribute__((ext_vector_type(2)));
typedef float v8f __attribute__((ext_vector_type(8)));

#define K_CODES 4096
#define D_DIM   32
#define HW      256     // 16*16
#define MB      4       // codebooks
#define NBATCH  8

// Outputs (concatenated flat, return order): sample, code, one_hot, logit
#define SAMPLE_N (NBATCH * MB * HW * K_CODES)   // 33554432
#define CODE_N   (NBATCH * MB * HW)             // 8192

// Deterministic gumbel noise keyed on flat logit element index (PCG-style hash).
__device__ __forceinline__ float gumbel_of(unsigned e) {
    unsigned s = e * 747796405u + 2891336453u;
    unsigned w = ((s >> ((s >> 28) + 4u)) ^ s) * 277803737u;
    w = (w >> 22) ^ w;
    float u = ((w >> 8) + 0.5f) * (1.0f / 16777216.0f);   // (0,1)
    const float eps = 1.1920929e-7f;
    u = fminf(fmaxf(u, eps), 1.0f - eps);
    return -__logf(-__logf(u));
}

__global__ __launch_bounds__(256)
void _multiCodebookQuantization_kernel(const float* __restrict__ x,
                                       const float* __restrict__ cb,
                                       const float* __restrict__ temp,
                                       float* __restrict__ out_sample,
                                       float* __restrict__ out_code,
                                       float* __restrict__ out_onehot,
                                       float* __restrict__ out_logit)
{
    const int wg   = blockIdx.x;        // 0..511: [m(4)][Mtile(128)]
    const int m    = wg >> 7;
    const int mt   = wg & 127;
    const int tid  = threadIdx.x;       // 0..255 (8 wave32s)
    const int lane = tid & 31;
    const int wave = tid >> 5;

    const int R0  = mt * 16;            // row base within this codebook's 2048 rows
    const int n   = R0 >> 8;            // batch index (16 rows never cross n)
    const int hw0 = R0 & 255;
    // contiguous 16*4096 output chunk base for this WG
    const int chunkBase = (n * (MB * HW) + m * HW + hw0) * K_CODES;

    __shared__ float s_x2[16];
    __shared__ float s_bLv[16][8];  __shared__ int s_bLi[16][8];
    __shared__ float s_bGv[16][8];  __shared__ int s_bGi[16][8];
    __shared__ int   s_finL[16];    __shared__ int s_finG[16];

    const int row  = lane & 15;         // A-matrix row / B-matrix column within tile
    const int half = lane >> 4;         // fp32 WMMA: lanes 16..31 hold K+2 slice
    const int koff = half * 2;

    const float tmax  = fmaxf(temp[m], 1e-6f);
    const float scale = tmax * (1.0f / 64.0f);   // 1/sqrt(4096)

    // ---- A fragments: 16 rows x K=32, per-lane slice; also partial ||x||^2 ----
    // x flat: (n*128 + m*32 + dd)*256 + hw
    const int baseA = n * (128 * HW) + m * (D_DIM * HW) + (hw0 + row);
    v2f av[8];
    float x2p = 0.f;
#pragma unroll
    for (int kk = 0; kk < 8; ++kk) {
        const int dd = 4 * kk + koff;
        const float a0 = x[baseA + dd * HW];
        const float a1 = x[baseA + (dd + 1) * HW];
        av[kk].x = a0; av[kk].y = a1;
        x2p += a0 * a0 + a1 * a1;
    }
    // lanes l and l^16 hold complementary halves of K -> full ||x||^2 per row
    const float x2full = x2p + __shfl_xor(x2p, 16, 32);
    if (wave == 0 && half == 0) s_x2[row] = x2full;
    __syncthreads();

    float x2r[8];
#pragma unroll
    for (int v = 0; v < 8; ++v) x2r[v] = s_x2[v + half * 8];

    // ---- per-lane argmax trackers (8 D-rows each) ----
    float bLv[8], bGv[8]; int bLi[8], bGi[8];
#pragma unroll
    for (int v = 0; v < 8; ++v) { bLv[v] = -INFINITY; bGv[v] = -INFINITY; bLi[v] = 0; bGi[v] = 0; }

    const int cbBase  = m * (K_CODES * D_DIM);
    const int colLane = lane & 15;

    // warm the L2 path for this wave's codebook slice
    __builtin_prefetch(cb + cbBase + (wave * 512 + colLane) * D_DIM, 0, 1);

    for (int t = 0; t < 32; ++t) {
        const int col = wave * 512 + t * 16 + colLane;
        const float* __restrict__ cbp = cb + cbBase + col * D_DIM;

        v8f c = {};          // accumulator (inline 0 C on first WMMA chain step)
        float c2p = 0.f;
#pragma unroll
        for (int kk = 0; kk < 8; ++kk) {
            const int dd = 4 * kk + koff;
            v2f b;
            b.x = cbp[dd];
            b.y = cbp[dd + 1];
            c2p += b.x * b.x + b.y * b.y;
            // D = A(16x4) * B(4x16) + C, chained over K=32
            c = __builtin_amdgcn_wmma_f32_16x16x4_f32(
                    /*neg_a=*/false, av[kk],
                    /*neg_b=*/false, b,
                    /*c_mod=*/(short)0, c,
                    /*reuse_a=*/false, /*reuse_b=*/false);
        }
        const float c2 = c2p + __shfl_xor(c2p, 16, 32);   // full ||codeword||^2

#pragma unroll
        for (int v = 0; v < 8; ++v) {
            const int rowD = v + half * 8;
            // logit = -(x2 + c2 - 2*inter)/sqrt(k) * max(temp,eps)
            const float val = (2.0f * c[v] - x2r[v] - c2) * scale;
            const int eidx = chunkBase + rowD * K_CODES + col;
            out_logit[eidx] = val;
            if (val > bLv[v]) { bLv[v] = val; bLi[v] = col; }
            const float vg = val + gumbel_of((unsigned)eidx);
            if (vg > bGv[v]) { bGv[v] = vg; bGi[v] = col; }
        }
    }

    // ---- reduce argmax across the 16 lanes sharing each row set ----
#pragma unroll
    for (int off = 8; off >= 1; off >>= 1) {
#pragma unroll
        for (int v = 0; v < 8; ++v) {
            float ov = __shfl_xor(bLv[v], off, 32);
            int   oi = __shfl_xor(bLi[v], off, 32);
            if (ov > bLv[v] || (ov == bLv[v] && oi < bLi[v])) { bLv[v] = ov; bLi[v] = oi; }
            ov = __shfl_xor(bGv[v], off, 32);
            oi = __shfl_xor(bGi[v], off, 32);
            if (ov > bGv[v] || (ov == bGv[v] && oi < bGi[v])) { bGv[v] = ov; bGi[v] = oi; }
        }
    }
    if (colLane == 0) {
#pragma unroll
        for (int v = 0; v < 8; ++v) {
            const int rowD = v + half * 8;
            s_bLv[rowD][wave] = bLv[v]; s_bLi[rowD][wave] = bLi[v];
            s_bGv[rowD][wave] = bGv[v]; s_bGi[rowD][wave] = bGi[v];
        }
    }
    __syncthreads();

    // ---- final cross-wave reduction; emit code (cast to float) ----
    if (tid < 16) {
        float bv = s_bLv[tid][0]; int bi = s_bLi[tid][0];
        float gv = s_bGv[tid][0]; int gi = s_bGi[tid][0];
#pragma unroll
        for (int wv = 1; wv < 8; ++wv) {
            float v2 = s_bLv[tid][wv]; int i2 = s_bLi[tid][wv];
            if (v2 > bv || (v2 == bv && i2 < bi)) { bv = v2; bi = i2; }
            v2 = s_bGv[tid][wv]; i2 = s_bGi[tid][wv];
            if (v2 > gv || (v2 == gv && i2 < gi)) { gv = v2; gi = i2; }
        }
        s_finL[tid] = bi;
        s_finG[tid] = gi;
        out_code[chunkBase / K_CODES + tid] = (float)bi;
    }
    __syncthreads();

    // ---- one-hot outputs: zero-fill fused with the 1.0 injection (float4) ----
    float4* __restrict__ sp = (float4*)(out_sample + chunkBase);
    float4* __restrict__ op = (float4*)(out_onehot + chunkBase);
    const float4 z = make_float4(0.f, 0.f, 0.f, 0.f);
    for (int i = tid; i < (16 * K_CODES) / 4; i += 256) {
        const int r  = i >> 10;             // 1024 float4 per row
        const int c0 = (i & 1023) * 4;
        float4 s = z, o = z;
        const int fg = s_finG[r] - c0;
        if ((unsigned)fg < 4u) ((float*)&s)[fg] = 1.0f;   // sample = one_hot(argmax(logit+gumbel))
        const int fl = s_finL[r] - c0;
        if ((unsigned)fl < 4u) ((float*)&o)[fl] = 1.0f;   // one_hot = one_hot(argmax(logit))
        sp[i] = s;
        op[i] = o;
    }
}

extern "C" void kernel_launch(void* const* d_in, const int* in_sizes, int n_in,
                              void* d_out, int out_size, void* d_ws, size_t ws_size,
                              hipStream_t stream) {
    const float* x    = (const float*)d_in[0];   // (8,128,16,16)
    const float* cb   = (const float*)d_in[1];   // (4,4096,32)
    const float* temp = (const float*)d_in[2];   // (4,1,1,1)

    float* out        = (float*)d_out;
    float* out_sample = out;
    float* out_code   = out + SAMPLE_N;
    float* out_onehot = out + SAMPLE_N + CODE_N;
    float* out_logit  = out + 2 * SAMPLE_N + CODE_N;

    // 4 codebooks * 128 M-tiles = 512 workgroups, 256 threads (8 wave32s) each
    _multiCodebookQuantization_kernel<<<512, 256, 0, stream>>>(
        x, cb, temp, out_sample, out_code, out_onehot, out_logit);
}